// ResidualGCN_45518063403398
// MI455X (gfx1250) — compile-verified
//
#include <hip/hip_runtime.h>
#include <hip/hip_bf16.h>

typedef float v2f __attribute__((ext_vector_type(2)));
typedef float v8f __attribute__((ext_vector_type(8)));

#define WAVES_PER_BLOCK 8

// ---------------- degree / normalization ----------------

__global__ void __launch_bounds__(256)
k_deg_init(float* __restrict__ deg, int n) {
    int i = blockIdx.x * 256 + threadIdx.x;
    if (i < n) deg[i] = 1.0f;  // self-loop contributes 1 to every node
}

__global__ void __launch_bounds__(256)
k_deg_edges(const int* __restrict__ col, float* __restrict__ deg, int e) {
    int i = blockIdx.x * 256 + threadIdx.x;
    if (i < e) atomicAdd(&deg[col[i]], 1.0f);
}

__global__ void __launch_bounds__(256)
k_inv_sqrt(float* __restrict__ deg, int n) {
    int i = blockIdx.x * 256 + threadIdx.x;
    if (i < n) deg[i] = rsqrtf(deg[i]);  // deg >= 1 always
}

// ---------------- layer 1 GEMM: xW1 = x@W1, id1 = x@t1_W, agg1 = xW1*inv^2 ----------------
// One wave computes a 16-row tile; K=128 in steps of 4 via v_wmma_f32_16x16x4_f32.

__global__ void __launch_bounds__(256)
k_gemm1(const float* __restrict__ x, const float* __restrict__ W1,
        const float* __restrict__ T1, const float* __restrict__ inv,
        float* __restrict__ xW1, float* __restrict__ agg1,
        float* __restrict__ id1, int nTiles) {
    __shared__ float lw[128 * 16];
    __shared__ float lt[128 * 16];
    for (int i = threadIdx.x; i < 128 * 16; i += 256) {
        lw[i] = W1[i];
        lt[i] = T1[i];
    }
    __syncthreads();

    int wave  = threadIdx.x >> 5;
    int lane  = threadIdx.x & 31;
    int tileM = blockIdx.x * WAVES_PER_BLOCK + wave;
    if (tileM >= nTiles) return;          // wave-uniform exit: EXEC stays all-1s for WMMA

    int m0    = tileM * 16;
    int mrow  = m0 + (lane & 15);         // A: lanes 0-15 / 16-31 both map M=0..15
    int khalf = (lane >> 4) << 1;         // A/B: low half K=k,k+1 ; high half K=k+2,k+3
    int n     = lane & 15;

    v8f c0 = {};
    v8f c1 = {};
#pragma unroll 8
    for (int kk = 0; kk < 128; kk += 4) {
        int kb = kk + khalf;
        v2f a;
        a.x = x[(size_t)mrow * 128 + kb];
        a.y = x[(size_t)mrow * 128 + kb + 1];
        v2f b;
        b.x = lw[kb * 16 + n];
        b.y = lw[(kb + 1) * 16 + n];
        c0 = __builtin_amdgcn_wmma_f32_16x16x4_f32(false, a, false, b, (short)0, c0, false, false);
        v2f t;
        t.x = lt[kb * 16 + n];
        t.y = lt[(kb + 1) * 16 + n];
        c1 = __builtin_amdgcn_wmma_f32_16x16x4_f32(false, a, false, t, (short)0, c1, false, false);
    }

    int rbase = m0 + ((lane >> 4) << 3);  // C: VGPR j -> row m0+j (lanes 0-15) / m0+8+j (16-31)
#pragma unroll
    for (int j = 0; j < 8; j++) {
        int row  = rbase + j;
        float iv = inv[row];
        float v  = c0[j];
        xW1[(size_t)row * 16 + n]  = v;
        agg1[(size_t)row * 16 + n] = v * iv * iv;   // self-loop term folded in
        id1[(size_t)row * 16 + n]  = c1[j];
    }
}

// ---------------- layer 1 scatter: agg1[col] += xW1[row]*inv[row]*inv[col] ----------------

__global__ void __launch_bounds__(256)
k_scatter1(const int* __restrict__ row, const int* __restrict__ col,
           const float* __restrict__ inv, const float* __restrict__ xW1,
           float* __restrict__ agg1, int e) {
    int i = blockIdx.x * 256 + threadIdx.x;
    if (i >= e) return;
    int r = row[i];
    int c = col[i];
    float w = inv[r] * inv[c];
    const float4* src = (const float4*)(xW1 + (size_t)r * 16);
    float* dst = agg1 + (size_t)c * 16;
#pragma unroll
    for (int q = 0; q < 4; q++) {
        float4 v = src[q];
        atomicAdd(dst + q * 4 + 0, v.x * w);
        atomicAdd(dst + q * 4 + 1, v.y * w);
        atomicAdd(dst + q * 4 + 2, v.z * w);
        atomicAdd(dst + q * 4 + 3, v.w * w);
    }
}

// ---------------- layer 1 finalize: h = relu(agg1 + b1) + id1 + t1_b (h in-place in agg1) ----

__global__ void __launch_bounds__(256)
k_fin1(float* __restrict__ agg1, const float* __restrict__ id1,
       const float* __restrict__ b1, const float* __restrict__ t1b, int total) {
    int i = blockIdx.x * 256 + threadIdx.x;
    if (i >= total) return;
    int ch = i & 15;
    float v = agg1[i] + b1[ch];
    v = v > 0.0f ? v : 0.0f;
    agg1[i] = v + id1[i] + t1b[ch];
}

// ---------------- layer 2 GEMM: hW2 = h@W2, id2 = h@t2_W, agg2(d_out) = hW2*inv^2 --------
// K=16, N=64 for each weight -> 8 accumulator tiles per wave.

__global__ void __launch_bounds__(256)
k_gemm2(const float* __restrict__ h, const float* __restrict__ W2,
        const float* __restrict__ T2, const float* __restrict__ inv,
        float* __restrict__ hW2, float* __restrict__ agg2,
        float* __restrict__ id2, int nTiles) {
    __shared__ float lw[16 * 64];
    __shared__ float lt[16 * 64];
    for (int i = threadIdx.x; i < 16 * 64; i += 256) {
        lw[i] = W2[i];
        lt[i] = T2[i];
    }
    __syncthreads();

    int wave  = threadIdx.x >> 5;
    int lane  = threadIdx.x & 31;
    int tileM = blockIdx.x * WAVES_PER_BLOCK + wave;
    if (tileM >= nTiles) return;

    int m0    = tileM * 16;
    int mrow  = m0 + (lane & 15);
    int khalf = (lane >> 4) << 1;
    int n     = lane & 15;

    v8f c0[4] = {{}, {}, {}, {}};
    v8f c1[4] = {{}, {}, {}, {}};
#pragma unroll
    for (int kk = 0; kk < 16; kk += 4) {
        int kb = kk + khalf;
        v2f a;
        a.x = h[(size_t)mrow * 16 + kb];
        a.y = h[(size_t)mrow * 16 + kb + 1];
#pragma unroll
        for (int nt = 0; nt < 4; nt++) {
            v2f b;
            b.x = lw[kb * 64 + nt * 16 + n];
            b.y = lw[(kb + 1) * 64 + nt * 16 + n];
            c0[nt] = __builtin_amdgcn_wmma_f32_16x16x4_f32(false, a, false, b, (short)0, c0[nt], false, false);
            v2f t;
            t.x = lt[kb * 64 + nt * 16 + n];
            t.y = lt[(kb + 1) * 64 + nt * 16 + n];
            c1[nt] = __builtin_amdgcn_wmma_f32_16x16x4_f32(false, a, false, t, (short)0, c1[nt], false, false);
        }
    }

    int rbase = m0 + ((lane >> 4) << 3);
#pragma unroll
    for (int nt = 0; nt < 4; nt++) {
#pragma unroll
        for (int j = 0; j < 8; j++) {
            int row   = rbase + j;
            int cc    = nt * 16 + n;
            float iv  = inv[row];
            float v   = c0[nt][j];
            hW2[(size_t)row * 64 + cc]  = v;
            agg2[(size_t)row * 64 + cc] = v * iv * iv;
            id2[(size_t)row * 64 + cc]  = c1[nt][j];
        }
    }
}

// ---------------- layer 2 scatter: 4 threads per edge, 16 channels each ----------------

__global__ void __launch_bounds__(256)
k_scatter2(const int* __restrict__ row, const int* __restrict__ col,
           const float* __restrict__ inv, const float* __restrict__ hW2,
           float* __restrict__ agg2, int e) {
    int t = blockIdx.x * 256 + threadIdx.x;
    int i = t >> 2;
    if (i >= e) return;
    int part = (t & 3) * 16;
    int r = row[i];
    int c = col[i];
    float w = inv[r] * inv[c];
    const float4* src = (const float4*)(hW2 + (size_t)r * 64 + part);
    float* dst = agg2 + (size_t)c * 64 + part;
#pragma unroll
    for (int q = 0; q < 4; q++) {
        float4 v = src[q];
        atomicAdd(dst + q * 4 + 0, v.x * w);
        atomicAdd(dst + q * 4 + 1, v.y * w);
        atomicAdd(dst + q * 4 + 2, v.z * w);
        atomicAdd(dst + q * 4 + 3, v.w * w);
    }
}

// ---------------- layer 2 finalize (in place in d_out) ----------------

__global__ void __launch_bounds__(256)
k_fin2(float* __restrict__ out, const float* __restrict__ id2,
       const float* __restrict__ b2, const float* __restrict__ t2b, int total) {
    int i = blockIdx.x * 256 + threadIdx.x;
    if (i >= total) return;
    int ch = i & 63;
    float v = out[i] + b2[ch];
    v = v > 0.0f ? v : 0.0f;
    out[i] = v + id2[i] + t2b[ch];
}

// ---------------- launch ----------------

extern "C" void kernel_launch(void* const* d_in, const int* in_sizes, int n_in,
                              void* d_out, int out_size, void* d_ws, size_t ws_size,
                              hipStream_t stream) {
    const float* x   = (const float*)d_in[0];
    const int*   ei  = (const int*)d_in[1];   // edge_index [2, E]
    const float* W1  = (const float*)d_in[2];
    const float* b1  = (const float*)d_in[3];
    const float* W2  = (const float*)d_in[4];
    const float* b2  = (const float*)d_in[5];
    const float* T1  = (const float*)d_in[6];
    const float* t1b = (const float*)d_in[7];
    const float* T2  = (const float*)d_in[8];
    const float* t2b = (const float*)d_in[9];

    const int N = in_sizes[0] / 128;
    const int E = in_sizes[1] / 2;
    const int* row = ei;
    const int* col = ei + E;

    float* ws   = (float*)d_ws;
    float* inv  = ws;                        // N        (deg, then deg^-1/2)
    float* xW1  = inv + (size_t)N;           // 16N
    float* agg1 = xW1 + (size_t)16 * N;      // 16N      (becomes h after fin1)
    float* id1  = agg1 + (size_t)16 * N;     // 16N
    float* hW2  = id1 + (size_t)16 * N;      // 64N
    float* id2  = hW2 + (size_t)64 * N;      // 64N
    float* out  = (float*)d_out;             // agg2 lives directly in d_out

    const int nTiles = N / 16;               // N = 100000 -> 6250 full tiles

    k_deg_init<<<(N + 255) / 256, 256, 0, stream>>>(inv, N);
    k_deg_edges<<<(E + 255) / 256, 256, 0, stream>>>(col, inv, E);
    k_inv_sqrt<<<(N + 255) / 256, 256, 0, stream>>>(inv, N);

    k_gemm1<<<(nTiles + WAVES_PER_BLOCK - 1) / WAVES_PER_BLOCK, 256, 0, stream>>>(
        x, W1, T1, inv, xW1, agg1, id1, nTiles);
    k_scatter1<<<(E + 255) / 256, 256, 0, stream>>>(row, col, inv, xW1, agg1, E);
    k_fin1<<<(N * 16 + 255) / 256, 256, 0, stream>>>(agg1, id1, b1, t1b, N * 16);

    k_gemm2<<<(nTiles + WAVES_PER_BLOCK - 1) / WAVES_PER_BLOCK, 256, 0, stream>>>(
        agg1 /*h*/, W2, T2, inv, hW2, out, id2, nTiles);
    k_scatter2<<<((E * 4) + 255) / 256, 256, 0, stream>>>(row, col, inv, hW2, out, E);
    k_fin2<<<(N * 64 + 255) / 256, 256, 0, stream>>>(out, id2, b2, t2b, N * 64);
}